// XnorWeight_52913997086768
// MI455X (gfx1250) — compile-verified
//
#include <hip/hip_runtime.h>
#include <hip/hip_bf16.h>
#include <stdint.h>

// XnorWeight: out[b,o,0,p] = float( ~(x[b,p] ^ W[o,0,p]) & 0xFF )
// Shapes: x (128, 512) int32, W (4096, 1, 512) int32, out (128,4096,1,512) -> float.
// Store-bandwidth bound: 1 GiB of float output @ 23.3 TB/s ~= 46 us floor.
// CDNA5 paths used: TDM tensor_load_to_lds (TENSORcnt) for operand staging,
// ds_load_b128 from LDS, non-temporal global_store_b128 for the streamed output.

#define BATCH   128
#define OUTF    4096
#define PACKED  512           // packed_in = 4096/8
#define OTILE   16            // output rows per block
#define BLOCK   (OTILE * 32)  // 16 wave32 waves

typedef __attribute__((ext_vector_type(4))) unsigned int u32x4;
typedef __attribute__((ext_vector_type(8))) int          i32x8;
typedef __attribute__((ext_vector_type(4))) int          i32x4;
typedef __attribute__((ext_vector_type(4))) float        f32x4;

// ---- Tensor DMA Descriptor builders (CDNA5 ISA) ----

// Group 0 (128 bits): [1:0]=count=1 (valid), [63:32]=lds_addr (bytes),
// [120:64]=global_addr (bytes), [127:126]=type=2 ("image").
__device__ __forceinline__ u32x4 tdm_group0(uint32_t lds_addr, uint64_t gaddr) {
  u32x4 g;
  g.x = 0x1u;
  g.y = lds_addr;
  g.z = (uint32_t)gaddr;
  g.w = ((uint32_t)(gaddr >> 32) & 0x01FFFFFFu) | (2u << 30);
  return g;
}

// Group 1 (256 bits): [17:16]=data_size (2 -> 4 bytes), [79:48]=tensor_dim0,
// [111:80]=tensor_dim1, [127:112]=tile_dim0, [143:128]=tile_dim1,
// [159:144]=tile_dim2 (0=unused), [207:160]=tensor_dim0_stride,
// [255:208]=tensor_dim1_stride (unused here). workgroup_mask must be 0 (no cluster).
__device__ __forceinline__ i32x8 tdm_group1(uint32_t tdim0, uint32_t tdim1,
                                            uint32_t tile0, uint32_t tile1,
                                            uint32_t d0_stride) {
  i32x8 g;
  g[0] = (int)(2u << 16);                                            // data_size = 4B
  g[1] = (int)((tdim0 & 0xFFFFu) << 16);                             // tensor_dim0 lo
  g[2] = (int)(((tdim0 >> 16) & 0xFFFFu) | ((tdim1 & 0xFFFFu) << 16));
  g[3] = (int)(((tdim1 >> 16) & 0xFFFFu) | ((tile0 & 0xFFFFu) << 16));
  g[4] = (int)(tile1 & 0xFFFFu);                                     // tile_dim2 = 0
  g[5] = (int)d0_stride;                                             // stride0[31:0]
  g[6] = 0;                                                          // stride0 hi / stride1 lo
  g[7] = 0;                                                          // stride1 hi
  return g;
}

__global__ __launch_bounds__(BLOCK)
void XnorWeight_52913997086768_kernel(const int* __restrict__ x,
                                      const int* __restrict__ W,
                                      float* __restrict__ out) {
  __shared__ uint32_t sX[PACKED];           // 2 KiB: x row for this batch b
  __shared__ uint32_t sW[OTILE * PACKED];   // 32 KiB: 16 output rows of W

  const int b      = blockIdx.y;
  const int o_base = blockIdx.x * OTILE;

  // Wave 0 issues both TDM loads (tensor ops are wave-level; EXEC ignored),
  // waits TENSORcnt==0, then the whole workgroup barriers.
  if ((threadIdx.x >> 5) == 0) {
    const uint32_t ldsX = (uint32_t)(uintptr_t)(&sX[0]);
    const uint32_t ldsW = (uint32_t)(uintptr_t)(&sW[0]);
    const uint64_t gaX  = (uint64_t)(uintptr_t)(x + (size_t)b * PACKED);
    const uint64_t gaW  = (uint64_t)(uintptr_t)(W + (size_t)o_base * PACKED);
    const i32x4 zero4 = {0, 0, 0, 0};
    const i32x8 zero8 = {0, 0, 0, 0, 0, 0, 0, 0};

    // x row: 1D tile of 512 dwords
    __builtin_amdgcn_tensor_load_to_lds(
        tdm_group0(ldsX, gaX),
        tdm_group1(PACKED, 1, PACKED, 1, PACKED),
        zero4, zero4, zero8, 0);
    // W tile: 2D tile, 16 rows x 512 dwords, row stride 512 dwords
    __builtin_amdgcn_tensor_load_to_lds(
        tdm_group0(ldsW, gaW),
        tdm_group1(PACKED, OUTF, PACKED, OTILE, PACKED),
        zero4, zero4, zero8, 0);

    __builtin_amdgcn_s_wait_tensorcnt(0);
  }
  __syncthreads();

  const int o_local = threadIdx.x >> 5;   // wave -> output row within tile
  const int lane    = threadIdx.x & 31;
  const size_t row  = ((size_t)b * OUTF + (size_t)(o_base + o_local)) * PACKED;

  const i32x4* __restrict__ vX = reinterpret_cast<const i32x4*>(&sX[0]);
  const i32x4* __restrict__ vW = reinterpret_cast<const i32x4*>(&sW[o_local * PACKED]);

#pragma unroll
  for (int i = 0; i < 4; ++i) {
    const int p4 = lane + i * 32;         // float4 group index, 0..127
    const i32x4 xv = vX[p4];              // ds_load_b128
    const i32x4 wv = vW[p4];              // ds_load_b128
    // ~(a^b) & 0xFF == (a^b)^0xFF for byte-range inputs -> v_xor3-friendly
    f32x4 f;
    f.x = (float)((xv.x ^ wv.x) ^ 0xFF);
    f.y = (float)((xv.y ^ wv.y) ^ 0xFF);
    f.z = (float)((xv.z ^ wv.z) ^ 0xFF);
    f.w = (float)((xv.w ^ wv.w) ^ 0xFF);
    // Streaming 1 GiB output: non-temporal b128 store, keep x/W hot in L2.
    __builtin_nontemporal_store(f, reinterpret_cast<f32x4*>(out + row + (size_t)p4 * 4));
  }
}

extern "C" void kernel_launch(void* const* d_in, const int* in_sizes, int n_in,
                              void* d_out, int out_size, void* d_ws, size_t ws_size,
                              hipStream_t stream) {
  (void)in_sizes; (void)n_in; (void)d_ws; (void)ws_size; (void)out_size;
  const int* x = (const int*)d_in[0];   // (128, 512) int32
  const int* W = (const int*)d_in[1];   // (4096, 1, 512) int32
  float* out   = (float*)d_out;         // (128, 4096, 1, 512) as float

  dim3 grid(OUTF / OTILE, BATCH);       // (256, 128)
  XnorWeight_52913997086768_kernel<<<grid, BLOCK, 0, stream>>>(x, W, out);
}